// Cropping_38543036514722
// MI455X (gfx1250) — compile-verified
//
#include <hip/hip_runtime.h>
#include <hip/hip_bf16.h>
#include <stdint.h>

#define OUT_SZ  50
#define B_DIM   8
#define C_DIM   3
#define H_DIM   1024
#define W_DIM   1024
#define N_BOX   128
#define SLABS   (B_DIM * C_DIM)
#define THREADS 256

__global__ __launch_bounds__(THREADS)
void crop_resize_kernel(const float* __restrict__ x,
                        const float* __restrict__ bbox,
                        float* __restrict__ out)
{
    __shared__ float sbox[4];
    __shared__ int   sx0[OUT_SZ], sx1[OUT_SZ], sy0[OUT_SZ], sy1[OUT_SZ];
    __shared__ float swx[OUT_SZ], swy[OUT_SZ];

    const int bid  = blockIdx.x;            // bid = n*24 + (b*C + c)
    const int n    = bid / SLABS;
    const int slab = bid - n * SLABS;       // b*C + c
    const int tid  = threadIdx.x;

    // ---- CDNA5 async global->LDS broadcast of this box's 4 floats ----
    // GLOBAL_LOAD_ASYNC_TO_LDS_B128 (ASYNCcnt), lane 0 only; the LDS dest
    // offset is the low 32 bits of the generic shared-pointer address.
    if (tid == 0) {
        uint32_t lds_off     = (uint32_t)(uintptr_t)(&sbox[0]);
        const float* gsrc    = bbox + (size_t)n * 4;
        asm volatile("global_load_async_to_lds_b128 %0, %1, off"
                     :: "v"(lds_off), "v"(gsrc)
                     : "memory");
        asm volatile("s_wait_asynccnt 0" ::: "memory");
    }
    __syncthreads();

    // ---- Build the 50-entry source-coordinate tables (shared by all 24 slabs) ----
    if (tid < OUT_SZ) {
        const float cx = sbox[0] * (float)W_DIM;
        const float cy = sbox[1] * (float)H_DIM;
        const float bw = sbox[2] * (float)W_DIM;
        const float bh = sbox[3] * (float)H_DIM;

        int x1 = (int)floorf(cx - 0.5f * bw); x1 = min(max(x1, 0), W_DIM - 1);
        int x2 = (int)floorf(cx + 0.5f * bw); x2 = min(max(x2, 0), W_DIM);
        x2 = max(x2, x1 + 1);
        int y1 = (int)floorf(cy - 0.5f * bh); y1 = min(max(y1, 0), H_DIM - 1);
        int y2 = (int)floorf(cy + 0.5f * bh); y2 = min(max(y2, 0), H_DIM);
        y2 = max(y2, y1 + 1);

        const float t = (float)tid + 0.5f;

        {   // horizontal source coords
            const float fn = (float)(x2 - x1);
            float s = t * fn / (float)OUT_SZ - 0.5f;
            s = fminf(fmaxf(s, 0.0f), fn - 1.0f);
            const int i0 = (int)floorf(s);
            swx[tid] = s - (float)i0;
            sx0[tid] = x1 + i0;
            sx1[tid] = x1 + min(i0 + 1, x2 - x1 - 1);
        }
        {   // vertical source coords
            const float fn = (float)(y2 - y1);
            float s = t * fn / (float)OUT_SZ - 0.5f;
            s = fminf(fmaxf(s, 0.0f), fn - 1.0f);
            const int i0 = (int)floorf(s);
            swy[tid] = s - (float)i0;
            sy0[tid] = y1 + i0;
            sy1[tid] = y1 + min(i0 + 1, y2 - y1 - 1);
        }
    }
    __syncthreads();

    // ---- Gather + bilinear combine over the 50x50 slab ----
    const float* __restrict__ src = x   + (size_t)slab * (H_DIM * W_DIM);
    float*       __restrict__ dst = out + (size_t)bid  * (OUT_SZ * OUT_SZ);

    for (int idx = tid; idx < OUT_SZ * OUT_SZ; idx += THREADS) {
        const int   i  = idx / OUT_SZ;
        const int   j  = idx - i * OUT_SZ;
        const int   r0 = sy0[i], r1 = sy1[i];
        const float wy = swy[i];
        const int   c0 = sx0[j], c1 = sx1[j];
        const float wx = swx[j];

        const float* row0 = src + (size_t)r0 * W_DIM;
        const float* row1 = src + (size_t)r1 * W_DIM;

        // 4 independent gathers issued before any use -> max memory parallelism.
        // These stay regular-temporal: the input working set is reused across
        // the 24 slabs per box and overlapping boxes, and fits in the 192MB L2.
        const float v00 = row0[c0];
        const float v01 = row0[c1];
        const float v10 = row1[c0];
        const float v11 = row1[c1];

        const float top = (1.0f - wx) * v00 + wx * v01;
        const float bot = (1.0f - wx) * v10 + wx * v11;
        const float res = (1.0f - wy) * top + wy * bot;

        // Write-once output: non-temporal store (th:TH_STORE_NT) so the 30MB
        // of results don't evict the gather-reused input from L2.
        __builtin_nontemporal_store(res, &dst[idx]);
    }
}

extern "C" void kernel_launch(void* const* d_in, const int* in_sizes, int n_in,
                              void* d_out, int out_size, void* d_ws, size_t ws_size,
                              hipStream_t stream)
{
    const float* x    = (const float*)d_in[0];   // (8,3,1024,1024) f32
    const float* bbox = (const float*)d_in[1];   // (128,4) f32
    float* out = (float*)d_out;                  // (128,8,3,50,50) f32

    crop_resize_kernel<<<dim3(N_BOX * SLABS), dim3(THREADS), 0, stream>>>(x, bbox, out);

    (void)in_sizes; (void)n_in; (void)out_size; (void)d_ws; (void)ws_size;
}